// LegendreNN_32298154066273
// MI455X (gfx1250) — compile-verified
//
#include <hip/hip_runtime.h>

#define DIM 8
#define HID 128
#define WS  129   // weight LDS row stride (odd -> conflict-light transposed reads)
#define TS  132   // tangent buffer row stride (even -> aligned b64 A-fragments)
#define NW  4     // waves per workgroup
#define EPS 1e-6f

typedef __attribute__((ext_vector_type(2))) float v2f;
typedef __attribute__((ext_vector_type(8))) float v8f;

// intra-wave LDS ordering fence (same-wave DS ops are in-order; this stops the
// compiler from caching/reordering across the cross-lane handoff)
#define WAVE_SYNC() asm volatile("s_wait_dscnt 0" ::: "memory")

struct SMem {
  float wbuf[HID][WS];            // revolving W1 / W2 image        (66048 B)
  float w0[16][WS];               // W0                              (8256 B)
  float b0v[HID], b1v[HID], b2v[HID], w3v[HID];                   // (2048 B)
  float T1[NW][16][TS];           // per-wave tangent buffer A      (33792 B)
  float T2[NW][16][TS];           // per-wave tangent buffer B      (33792 B)
  float sv[NW][4][HID];           // 0:s0 1:s1 2:c2|a0 3:a1          (8192 B)
  float xq[NW][16];               // [q, qdot]                        (256 B)
  float h0[NW][HID], h1[NW][HID];                                 // (4096 B)
  float u2[NW][HID], u1[NW][HID], u0[NW][HID];                    // (6144 B)
  float misc[NW][16];             // dLdq etc.                        (256 B)
};                                 // total = 162880 B  (<= 160 KB)
static_assert(sizeof(SMem) <= 163840, "LDS budget");

__device__ inline float softplus_f(float z) {
  return fmaxf(z, 0.f) + log1pf(expf(-fabsf(z)));
}
__device__ inline float sigmoid_f(float z) { return 1.f / (1.f + expf(-z)); }

__device__ inline v8f wmma4(v2f a, v2f b, v8f c) {
  // D = A(16x4, f32) x B(4x16, f32) + C(16x16, f32)
  return __builtin_amdgcn_wmma_f32_16x16x4_f32(false, a, false, b, (short)0, c,
                                               false, false);
}

__device__ inline void load_big(float (*dst)[WS], const float* __restrict__ g) {
  for (int idx = threadIdx.x * 4; idx < HID * HID; idx += blockDim.x * 4) {
    float4 v = *(const float4*)(g + idx);
    int r = idx >> 7, c = idx & 127;
    dst[r][c] = v.x; dst[r][c + 1] = v.y; dst[r][c + 2] = v.z; dst[r][c + 3] = v.w;
  }
}

// acc(16x128) += Asrc(16x128, optional k-scale) @ B   (B = wbuf or wbuf^T)
template <bool BTRANS, bool ASCALE>
__device__ inline void gemm128(const float* Arow, int AS, const float* scale,
                               const float (*Bw)[WS], v8f acc[8], int fm, int fk) {
  for (int kb = 0; kb < HID; kb += 4) {
    int k0 = kb + fk;
    v2f a;
    a.x = Arow[fm * AS + k0];
    a.y = Arow[fm * AS + k0 + 1];
    if (ASCALE) { a.x *= scale[k0]; a.y *= scale[k0 + 1]; }
#pragma unroll
    for (int nt = 0; nt < 8; nt++) {
      int nb = nt * 16;
      v2f b;
      if (BTRANS) { b.x = Bw[nb + fm][k0];  b.y = Bw[nb + fm][k0 + 1]; }
      else        { b.x = Bw[k0][nb + fm];  b.y = Bw[k0 + 1][nb + fm]; }
      acc[nt] = wmma4(a, b, acc[nt]);
    }
  }
}

__global__ __launch_bounds__(NW * 32) void lnn_kernel(
    const float* __restrict__ q, const float* __restrict__ qd,
    const float* __restrict__ W0g, const float* __restrict__ b0g,
    const float* __restrict__ W1g, const float* __restrict__ b1g,
    const float* __restrict__ W2g, const float* __restrict__ b2g,
    const float* __restrict__ W3g, float* __restrict__ out, int B, int nbatch) {
  __shared__ SMem sm;
  const int tid = threadIdx.x;
  const int lane = tid & 31;
  const int wv = tid >> 5;
  const int fm = lane & 15;        // A row / B col / C col
  const int fk = (lane >> 4) * 2;  // A/B k-offset inside 4-wide k block
  const int ch = lane >> 4;        // C half (rows r / r+8)

  // ---- stage loop-invariant weights ----
  for (int idx = tid; idx < 16 * HID; idx += blockDim.x)
    sm.w0[idx >> 7][idx & 127] = W0g[idx];
  if (tid < HID) {
    sm.b0v[tid] = b0g[tid]; sm.b1v[tid] = b1g[tid];
    sm.b2v[tid] = b2g[tid]; sm.w3v[tid] = W3g[tid];
  }
  load_big(sm.wbuf, W1g);
  __syncthreads();

  for (int bat = blockIdx.x; bat < nbatch; bat += gridDim.x) {
    int smp = bat * NW + wv;
    bool valid = smp < B;  // wave-uniform -> EXEC stays all-ones inside

    if (valid) {
      // ================= phase 1 (wbuf = W1) =================
      if (lane < 8)       sm.xq[wv][lane] = q[smp * DIM + lane];
      else if (lane < 16) sm.xq[wv][lane] = qd[smp * DIM + (lane - 8)];
      WAVE_SYNC();
      // z0 / s0 / h0   (lane owns k = lane + 32*i)
      for (int i = 0; i < 4; i++) {
        int k = lane + 32 * i;
        float z = sm.b0v[k];
        for (int j = 0; j < 16; j++) z += sm.xq[wv][j] * sm.w0[j][k];
        sm.sv[wv][0][k] = sigmoid_f(z);
        sm.h0[wv][k] = softplus_f(z);
      }
      WAVE_SYNC();
      // z1 / s1 / h1
      {
        float z[4];
        for (int i = 0; i < 4; i++) z[i] = sm.b1v[lane + 32 * i];
        for (int j = 0; j < HID; j++) {
          float h = sm.h0[wv][j];
          for (int i = 0; i < 4; i++) z[i] += h * sm.wbuf[j][lane + 32 * i];
        }
        for (int i = 0; i < 4; i++) {
          int k = lane + 32 * i;
          sm.sv[wv][1][k] = sigmoid_f(z[i]);
          sm.h1[wv][k] = softplus_f(z[i]);
        }
      }
      WAVE_SYNC();
      // GEMM1: dZ1 = (s0 .* W0) @ W1 -> T1
      {
        v8f acc[8] = {};
        gemm128<false, true>(&sm.w0[0][0], WS, sm.sv[wv][0], sm.wbuf, acc, fm, fk);
#pragma unroll
        for (int nt = 0; nt < 8; nt++) {
          int k = nt * 16 + fm;
#pragma unroll
          for (int r = 0; r < 8; r++) sm.T1[wv][r + 8 * ch][k] = acc[nt][r];
        }
      }
    }
    __syncthreads();
    load_big(sm.wbuf, W2g);
    __syncthreads();

    if (valid) {
      // ================= phase 2 (wbuf = W2) =================
      // z2 -> s2, c2 = s2'(z2)*W3, u2 = s2*W3
      {
        float z[4];
        for (int i = 0; i < 4; i++) z[i] = sm.b2v[lane + 32 * i];
        for (int j = 0; j < HID; j++) {
          float h = sm.h1[wv][j];
          for (int i = 0; i < 4; i++) z[i] += h * sm.wbuf[j][lane + 32 * i];
        }
        for (int i = 0; i < 4; i++) {
          int k = lane + 32 * i;
          float s2 = sigmoid_f(z[i]);
          float w3 = sm.w3v[k];
          sm.sv[wv][2][k] = s2 * (1.f - s2) * w3;  // c2
          sm.u2[wv][k] = s2 * w3;                  // u2
        }
      }
      WAVE_SYNC();
      // u1b = W2 @ u2 ; a1 = s1'*u1b ; u1 = s1*u1b
      for (int i = 0; i < 4; i++) {
        int k = lane + 32 * i;
        float ub = 0.f;
        for (int j = 0; j < HID; j++) ub += sm.wbuf[k][j] * sm.u2[wv][j];
        float s1k = sm.sv[wv][1][k];
        sm.sv[wv][3][k] = s1k * (1.f - s1k) * ub;  // a1
        sm.u1[wv][k] = s1k * ub;
      }
      WAVE_SYNC();
      // GEMM2: dZ2 = (s1 .* dZ1) @ W2 -> T2 (raw)
      {
        v8f acc[8] = {};
        gemm128<false, true>(&sm.T1[wv][0][0], TS, sm.sv[wv][1], sm.wbuf, acc, fm, fk);
#pragma unroll
        for (int nt = 0; nt < 8; nt++) {
          int k = nt * 16 + fm;
#pragma unroll
          for (int r = 0; r < 8; r++) sm.T2[wv][r + 8 * ch][k] = acc[nt][r];
        }
      }
      WAVE_SYNC();
      // GEMM3: dU1b = (c2 .* dZ2) @ W2^T ; dU1 = a1.*dZ1 + s1.*dU1b -> T2
      {
        v8f acc[8] = {};
        gemm128<true, true>(&sm.T2[wv][0][0], TS, sm.sv[wv][2], sm.wbuf, acc, fm, fk);
        WAVE_SYNC();
#pragma unroll
        for (int nt = 0; nt < 8; nt++) {
          int k = nt * 16 + fm;
          float a1k = sm.sv[wv][3][k], s1k = sm.sv[wv][1][k];
#pragma unroll
          for (int r = 0; r < 8; r++) {
            int t = r + 8 * ch;
            sm.T2[wv][t][k] = a1k * sm.T1[wv][t][k] + s1k * acc[nt][r];
          }
        }
      }
    }
    __syncthreads();
    load_big(sm.wbuf, W1g);
    __syncthreads();

    if (valid) {
      // ================= phase 3 (wbuf = W1) =================
      // u0b = W1 @ u1 ; a0 = s0'*u0b (reuse slot 2) ; u0 = s0*u0b
      for (int i = 0; i < 4; i++) {
        int k = lane + 32 * i;
        float ub = 0.f;
        for (int j = 0; j < HID; j++) ub += sm.wbuf[k][j] * sm.u1[wv][j];
        float s0k = sm.sv[wv][0][k];
        sm.sv[wv][2][k] = s0k * (1.f - s0k) * ub;  // a0
        sm.u0[wv][k] = s0k * ub;
      }
      WAVE_SYNC();
      // dL/dq_i = sum_k W0[i,k]*u0[k], i<8 (split-k over 4 lane groups, u2 = scratch)
      {
        int gi = lane & 7, ks = (lane >> 3) * 32;
        float part = 0.f;
        for (int k = ks; k < ks + 32; k++) part += sm.w0[gi][k] * sm.u0[wv][k];
        sm.u2[wv][lane] = part;
        WAVE_SYNC();
        if (lane < 8)
          sm.misc[wv][lane] = sm.u2[wv][lane] + sm.u2[wv][lane + 8] +
                              sm.u2[wv][lane + 16] + sm.u2[wv][lane + 24];
        WAVE_SYNC();
      }
      // GEMM4: dU0b = dU1 @ W1^T ; dU0 = a0.*W0 + s0.*dU0b -> T1
      {
        v8f acc[8] = {};
        gemm128<true, false>(&sm.T2[wv][0][0], TS, nullptr, sm.wbuf, acc, fm, fk);
#pragma unroll
        for (int nt = 0; nt < 8; nt++) {
          int k = nt * 16 + fm;
          float a0k = sm.sv[wv][2][k], s0k = sm.sv[wv][0][k];
#pragma unroll
          for (int r = 0; r < 8; r++) {
            int t = r + 8 * ch;
            sm.T1[wv][t][k] = a0k * sm.w0[t][k] + s0k * acc[nt][r];
          }
        }
      }
      WAVE_SYNC();
      // GEMM5: dG = dU0 @ W0^T (16x16 Hessian) -> T2 region, stride 17
      {
        v8f a5 = {};
        for (int kb = 0; kb < HID; kb += 4) {
          int k0 = kb + fk;
          v2f a; a.x = sm.T1[wv][fm][k0]; a.y = sm.T1[wv][fm][k0 + 1];
          v2f b; b.x = sm.w0[fm][k0];     b.y = sm.w0[fm][k0 + 1];
          a5 = wmma4(a, b, a5);
        }
        float* dg = &sm.T2[wv][0][0];
#pragma unroll
        for (int r = 0; r < 8; r++) dg[(r + 8 * ch) * 17 + fm] = a5[r];
      }
      WAVE_SYNC();
      // ----- build (M + eps I | rhs) and solve (partial-pivot GE) -----
      {
        float* dg = &sm.T2[wv][0][0];   // H[t][i] at dg[t*17+i]
        float* A  = &sm.T1[wv][0][0];   // 8x9 augmented system
        for (int idx = lane; idx < 64; idx += 32) {
          int i = idx >> 3, j = idx & 7;
          A[i * 9 + j] = dg[(8 + i) * 17 + (8 + j)] + ((i == j) ? EPS : 0.f);
        }
        if (lane < 8) {
          float r = sm.misc[wv][lane];                       // dL/dq_i
          for (int j = 0; j < 8; j++)
            r -= dg[(8 + lane) * 17 + j] * sm.xq[wv][8 + j]; // - C q_dot
          A[lane * 9 + 8] = r;
        }
        WAVE_SYNC();
        for (int c = 0; c < 8; c++) {
          int p = c; float best = fabsf(A[c * 9 + c]);   // redundant scan, all lanes
          for (int rr = c + 1; rr < 8; rr++) {
            float v = fabsf(A[rr * 9 + c]);
            if (v > best) { best = v; p = rr; }
          }
          if (p != c && lane < 9) {
            float t = A[c * 9 + lane];
            A[c * 9 + lane] = A[p * 9 + lane];
            A[p * 9 + lane] = t;
          }
          WAVE_SYNC();
          float piv = A[c * 9 + c];
          for (int rr = c + 1; rr < 8; rr++) {
            float f = A[rr * 9 + c] / piv;
            if (lane < 9) A[rr * 9 + lane] -= f * A[c * 9 + lane];
          }
          WAVE_SYNC();
        }
        if (lane == 0) {
          float x[8];
          for (int i = 7; i >= 0; i--) {
            float s = A[i * 9 + 8];
            for (int j = i + 1; j < 8; j++) s -= A[i * 9 + j] * x[j];
            x[i] = s / A[i * 9 + i];
          }
          for (int i = 0; i < 8; i++) out[smp * DIM + i] = x[i];
        }
      }
    }
  }
}

extern "C" void kernel_launch(void* const* d_in, const int* in_sizes, int n_in,
                              void* d_out, int out_size, void* d_ws, size_t ws_size,
                              hipStream_t stream) {
  const float* q  = (const float*)d_in[0];
  const float* qd = (const float*)d_in[1];
  const float* W0 = (const float*)d_in[2];
  const float* b0 = (const float*)d_in[3];
  const float* W1 = (const float*)d_in[4];
  const float* b1 = (const float*)d_in[5];
  const float* W2 = (const float*)d_in[6];
  const float* b2 = (const float*)d_in[7];
  const float* W3 = (const float*)d_in[8];
  float* out = (float*)d_out;
  int B = in_sizes[0] / DIM;
  int nbatch = (B + NW - 1) / NW;
  int grid = nbatch < 2048 ? nbatch : 2048;
  lnn_kernel<<<grid, NW * 32, 0, stream>>>(q, qd, W0, b0, W1, b1, W2, b2, W3,
                                           out, B, nbatch);
  (void)d_ws; (void)ws_size; (void)n_in; (void)out_size;
}